// quantum_hidden_Markov_neural_network_61366492725505
// MI455X (gfx1250) — compile-verified
//
#include <hip/hip_runtime.h>
#include <stdint.h>

// ---------------------------------------------------------------------------
// Quantum hidden Markov network: rho' = sum_k K_k rho K_k^dagger, 8 layers.
// One workgroup per batch density matrix; full chain computed with rho held
// in LDS; all matmuls on V_WMMA_F32_16X16X4_F32 (fp32 precision preserved).
// Kraus operators are double-buffered in LDS and fetched with the CDNA5
// async global->LDS path (GLOBAL_LOAD_ASYNC_TO_LDS_B128 / ASYNCcnt).
// ---------------------------------------------------------------------------

typedef __attribute__((ext_vector_type(2))) float v2f;
typedef __attribute__((ext_vector_type(8))) float v8f;

#define DD   64      // matrix dimension
#define KOP  16      // Kraus operators per layer
#define NLAY 8       // layers
#define SP   68      // padded LDS row stride (floats): 68%4==0 (float4 ok),
                     // bank(row) = 4*row mod 64 -> column reads conflict-free
#define TILE (DD * SP)

__device__ __forceinline__ v8f wmma_f32(v2f a, v2f b, v8f c) {
    // D = A(16x4, f32) * B(4x16, f32) + C(16x16, f32)
    return __builtin_amdgcn_wmma_f32_16x16x4_f32(
        /*neg_a=*/false, a, /*neg_b=*/false, b,
        /*c_mod=*/(short)0, c, /*reuse_a=*/false, /*reuse_b=*/false);
}

// complex MAC on 16x16 tiles:
//   Cr += Ar*Br - Ai*Bi   (nai = -ai precomputed)
//   Ci += Ar*Bi + Ai*Br
__device__ __forceinline__ void cwmma(v8f& cr, v8f& ci,
                                      v2f ar, v2f ai, v2f nai,
                                      v2f br, v2f bi) {
    cr = wmma_f32(ar,  br, cr);
    cr = wmma_f32(nai, bi, cr);
    ci = wmma_f32(ar,  bi, ci);
    ci = wmma_f32(ai,  br, ci);
}

// Synchronous copy of one 64x64 fp32 matrix global->LDS (padded stride SP).
__device__ __forceinline__ void stage_to_lds(float* dst, const float* src, int tid) {
    const float4* s = (const float4*)src;
#pragma unroll
    for (int t = 0; t < 2; ++t) {
        int i4  = tid * 2 + t;          // 0..1023
        int row = i4 >> 4;              // 16 float4 per row
        int c4  = i4 & 15;
        float4 v = s[i4];
        *(float4*)&dst[row * SP + c4 * 4] = v;
    }
}

// Async copy of one 64x64 fp32 matrix global->LDS via the Tensor/async engine:
// each lane issues a 16-byte GLOBAL_LOAD_ASYNC_TO_LDS_B128 (GVS addressing:
// SGPR 64-bit base + 32-bit VGPR byte offset). Tracked by ASYNCcnt.
__device__ __forceinline__ void stage_to_lds_async(float* dst, const float* src, int tid) {
#pragma unroll
    for (int t = 0; t < 2; ++t) {
        int i4  = tid * 2 + t;          // 0..1023 float4 chunks
        int row = i4 >> 4;
        int c4  = i4 & 15;
        uint32_t lds_addr = (uint32_t)(uintptr_t)&dst[row * SP + c4 * 4]; // LDS byte addr
        uint32_t goff     = (uint32_t)(i4 * 16);                          // global byte off
        asm volatile("global_load_async_to_lds_b128 %0, %1, %2 offset:0"
                     :
                     : "v"(lds_addr), "v"(goff), "s"(src)
                     : "memory");
    }
}

__device__ __forceinline__ void wait_async0() {
    asm volatile("s_wait_asynccnt 0x0" ::: "memory");
}

__global__ void __launch_bounds__(512)
qhmm_kraus_chain_kernel(const float* __restrict__ state_re,
                        const float* __restrict__ state_im,
                        const float* __restrict__ kraus_re,
                        const float* __restrict__ kraus_im,
                        float* __restrict__ out, int Bn) {
    extern __shared__ float smem[];
    float* Rr  = smem;            // rho real   [64][SP]
    float* Ri  = Rr  + TILE;      // rho imag
    float* Tr  = Ri  + TILE;      // tmp = K_k @ rho, real
    float* Ti  = Tr  + TILE;      // tmp imag
    float* Kr0 = Ti  + TILE;      // K double buffer
    float* Ki0 = Kr0 + TILE;
    float* Kr1 = Ki0 + TILE;
    float* Ki1 = Kr1 + TILE;

    const int b    = blockIdx.x;
    const int tid  = threadIdx.x;
    const int lane = tid & 31;
    const int wave = tid >> 5;        // 0..15 -> 4x4 grid of 16x16 tiles
    const int half = lane >> 4;       // lane half selects K-pair / row group
    const int l16  = lane & 15;
    const int tm   = (wave >> 2) << 4;   // tile row origin
    const int tn   = (wave & 3)  << 4;   // tile col origin

    // stage initial rho (synchronous; unique per block)
    stage_to_lds(Rr, state_re + (size_t)b * DD * DD, tid);
    stage_to_lds(Ri, state_im + (size_t)b * DD * DD, tid);

    // prologue: kick off async fetch of K_{0,0} into buffer 0
    stage_to_lds_async(Kr0, kraus_re, tid);
    stage_to_lds_async(Ki0, kraus_im, tid);

    for (int l = 0; l < NLAY; ++l) {
        v8f accR = {};   // rho' tile accumulators (persist across k)
        v8f accI = {};

        for (int k = 0; k < KOP; ++k) {
            const int t = l * KOP + k;          // flattened pipeline index
            float* Kr = (t & 1) ? Kr1 : Kr0;    // current K buffer
            float* Ki = (t & 1) ? Ki1 : Ki0;

            wait_async0();       // my async fills of the current buffer done
            __syncthreads();     // everyone's fills done & visible; prior
                                 // readers of the other buffer / Tr/Ti done

            // issue async fetch of the NEXT Kraus operator into other buffer
            if (t + 1 < NLAY * KOP) {
                const int nt = t + 1;
                const size_t nb = ((size_t)(nt >> 4) * KOP + (nt & 15)) * DD * DD;
                float* KrN = (nt & 1) ? Kr1 : Kr0;
                float* KiN = (nt & 1) ? Ki1 : Ki0;
                stage_to_lds_async(KrN, kraus_re + nb, tid);
                stage_to_lds_async(KiN, kraus_im + nb, tid);
            }

            // ---- phase 1: tmp tile (tm,tn) of K_k @ rho ----
            v8f cR = {};
            v8f cI = {};
#pragma unroll 4
            for (int l0 = 0; l0 < DD; l0 += 4) {
                // A fragment: rows tm..tm+15 of K_k, K-cols l0..l0+3
                int arow = (tm + l16) * SP + l0 + 2 * half;   // even -> b64 loads
                v2f ar  = { Kr[arow], Kr[arow + 1] };
                v2f ai  = { Ki[arow], Ki[arow + 1] };
                v2f nai = { -ai.x, -ai.y };
                // B fragment: rows l0..l0+3 of rho, cols tn..tn+15
                int brow = (l0 + 2 * half) * SP + tn + l16;
                v2f br = { Rr[brow], Rr[brow + SP] };
                v2f bi = { Ri[brow], Ri[brow + SP] };
                cwmma(cR, cI, ar, ai, nai, br, bi);
            }
            // spill tmp tile to LDS (C layout: VGPR v -> row v + 8*half)
#pragma unroll
            for (int v = 0; v < 8; ++v) {
                int row = tm + v + 8 * half;
                Tr[row * SP + tn + l16] = cR[v];
                Ti[row * SP + tn + l16] = cI[v];
            }
            __syncthreads();   // tmp ready (also: everyone done reading rho)

            // ---- phase 2: acc += tmp @ K_k^dagger ----
            // C[i,m] = sum_l tmp[i,l] * conj(K[m,l])  => B[l][m] = conj(K[m][l])
#pragma unroll 4
            for (int l0 = 0; l0 < DD; l0 += 4) {
                int arow = (tm + l16) * SP + l0 + 2 * half;
                v2f ar  = { Tr[arow], Tr[arow + 1] };
                v2f ai  = { Ti[arow], Ti[arow + 1] };
                v2f nai = { -ai.x, -ai.y };
                int bidx = (tn + l16) * SP + l0 + 2 * half;   // K row (tn+l16)
                v2f br = {  Kr[bidx],  Kr[bidx + 1] };
                v2f bi = { -Ki[bidx], -Ki[bidx + 1] };        // conjugate
                cwmma(accR, accI, ar, ai, nai, br, bi);
            }
        } // k

        __syncthreads();   // all phase-1 readers of rho long done; safe to overwrite
#pragma unroll
        for (int v = 0; v < 8; ++v) {
            int row = tm + v + 8 * half;
            Rr[row * SP + tn + l16] = accR[v];
            Ri[row * SP + tn + l16] = accI[v];
        }
    } // layers

    __syncthreads();
    // output: [2, B, 64, 64] -> re at b*4096, im at (Bn + b)*4096
    {
        float4* ore = (float4*)(out + (size_t)b * DD * DD);
        float4* oim = (float4*)(out + (size_t)(Bn + b) * DD * DD);
#pragma unroll
        for (int t = 0; t < 2; ++t) {
            int i4  = tid * 2 + t;
            int row = i4 >> 4;
            int c4  = i4 & 15;
            ore[i4] = *(const float4*)&Rr[row * SP + c4 * 4];
            oim[i4] = *(const float4*)&Ri[row * SP + c4 * 4];
        }
    }
}

extern "C" void kernel_launch(void* const* d_in, const int* in_sizes, int n_in,
                              void* d_out, int out_size, void* d_ws, size_t ws_size,
                              hipStream_t stream) {
    const float* state_re = (const float*)d_in[0];
    const float* state_im = (const float*)d_in[1];
    const float* kraus_re = (const float*)d_in[2];
    const float* kraus_im = (const float*)d_in[3];
    float* out = (float*)d_out;

    const int Bn = in_sizes[0] / (DD * DD);        // 1024
    const size_t shmem = (size_t)8 * TILE * sizeof(float);   // ~136 KB

    qhmm_kraus_chain_kernel<<<Bn, 512, shmem, stream>>>(
        state_re, state_im, kraus_re, kraus_im, out, Bn);
}